// ENCODER_62663572849388
// MI455X (gfx1250) — compile-verified
//
#include <hip/hip_runtime.h>
#include <hip/hip_bf16.h>
#include <stdint.h>

#define N_NODES 50000

typedef __attribute__((ext_vector_type(2))) float v2f;
typedef __attribute__((ext_vector_type(8))) float v8f;

// ---------------------------------------------------------------------------
// degree / normalization
// ---------------------------------------------------------------------------
__global__ void zero_f32(float* __restrict__ p, int n) {
    int i = blockIdx.x * blockDim.x + threadIdx.x;
    if (i < n) p[i] = 0.0f;
}

__global__ void deg_count(const int* __restrict__ dst, int E, float* __restrict__ deg) {
    int i = blockIdx.x * blockDim.x + threadIdx.x;
    if (i < E) atomicAdd(&deg[dst[i]], 1.0f);
}

__global__ void make_dis(float* __restrict__ d, int n) {
    int i = blockIdx.x * blockDim.x + threadIdx.x;
    if (i < n) d[i] = rsqrtf(d[i] + 1.0f);   // deg includes self-loop (+1)
}

// ---------------------------------------------------------------------------
// GEMM (fp32 WMMA 16x16x4) fused with row-scaling by dis:
//   G[i,:] = (X @ W)[i,:] * dis[i];  A initialized to G (self-loop term)
// One wave per 16x16 output tile, 4 waves per block. M=50000=3125*16,
// N in {64,128}: all tiles full -> EXEC all-ones (WMMA requirement).
// K is a template parameter (64 or 128) -> fully unrolled straight-line
// chain of 16/32 v_wmma_f32_16x16x4_f32 with immediate-offset loads.
// ---------------------------------------------------------------------------
template <int K>
__global__ void gemm_scale_wmma(const float* __restrict__ X, const float* __restrict__ W,
                                const float* __restrict__ dis,
                                float* __restrict__ G, float* __restrict__ A,
                                int N) {
    const int wid  = threadIdx.x >> 5;
    const int tile = blockIdx.x * 4 + wid;
    const int ntn  = N >> 4;
    const int tm   = tile / ntn;
    const int tn   = tile % ntn;
    const int lane = threadIdx.x & 31;
    const int half = lane >> 4;     // 0: K-phase {0,1}; 1: K-phase {2,3}
    const int l16  = lane & 15;

    const float* __restrict__ xrow = X + (size_t)(tm * 16 + l16) * K; // A: M = l16
    const float* __restrict__ wcol = W + (size_t)(tn * 16 + l16);     // B: N = l16

    v8f c = {};
#pragma unroll
    for (int k0 = 0; k0 < K; k0 += 4) {
        const int ka = k0 + 2 * half;
        v2f a, b;
        a.x = xrow[ka];                 // A[m][ka]   (contiguous pair -> b64 load)
        a.y = xrow[ka + 1];             // A[m][ka+1]
        b.x = wcol[(size_t)ka * N];     // B[ka][n]
        b.y = wcol[(size_t)(ka + 1) * N];
        c = __builtin_amdgcn_wmma_f32_16x16x4_f32(
                /*neg_a=*/false, a, /*neg_b=*/false, b,
                /*c_mod=*/(short)0, c, /*reuse_a=*/false, /*reuse_b=*/false);
    }

    // per-row normalization scale (C/D layout: row M = v + 8*half)
    float dv[8];
#pragma unroll
    for (int v = 0; v < 8; ++v) dv[v] = dis[tm * 16 + v + 8 * half];

#pragma unroll
    for (int v = 0; v < 8; ++v) {
        const int row   = tm * 16 + v + 8 * half;
        const float val = c[v] * dv[v];
        const size_t o  = (size_t)row * N + (size_t)(tn * 16 + l16);
        G[o] = val;
        A[o] = val;   // accumulator starts at self-loop contribution g[i]
    }
}

// ---------------------------------------------------------------------------
// Edge scatter: A[dst,:] += G[src,:]   (L2-resident f32 atomics)
// One thread per (edge, feature); N is a power of two -> shift/mask indexing.
// ---------------------------------------------------------------------------
__global__ void edge_scatter(const int* __restrict__ src, const int* __restrict__ dst,
                             long long E, int nshift,
                             const float* __restrict__ G, float* __restrict__ A) {
    const long long idx = (long long)blockIdx.x * blockDim.x + threadIdx.x;
    const long long tot = E << nshift;
    if (idx >= tot) return;
    const int  N = 1 << nshift;
    const long long e = idx >> nshift;
    const int  f = (int)(idx & (N - 1));
    const int  s = src[e];
    const int  d = dst[e];
    atomicAdd(&A[(size_t)d * N + f], G[(size_t)s * N + f]);
}

// ---------------------------------------------------------------------------
// Finalize: out = relu(dis[i] * A[i,f] + b[f])
// ---------------------------------------------------------------------------
__global__ void finalize_relu(const float* __restrict__ A, const float* __restrict__ dis,
                              const float* __restrict__ bia, float* __restrict__ out,
                              int nshift, long long tot) {
    const long long idx = (long long)blockIdx.x * blockDim.x + threadIdx.x;
    if (idx >= tot) return;
    const int N = 1 << nshift;
    const int f = (int)(idx & (N - 1));
    const long long row = idx >> nshift;
    const float v = dis[row] * A[idx] + bia[f];
    out[idx] = v > 0.0f ? v : 0.0f;
}

// ---------------------------------------------------------------------------
// Host-side layer driver (all launches on `stream`, graph-capture safe)
// ---------------------------------------------------------------------------
static inline int ilog2i(int n) { return 31 - __builtin_clz((unsigned)n); }

static void run_layer(const float* xin, const float* W, const float* bia,
                      int K, int N, const int* src, const int* dst, long long E,
                      const float* dis, float* G, float* A, float* out,
                      hipStream_t stream) {
    const int tiles = (N_NODES / 16) * (N / 16);      // 12500 or 25000, both %4==0
    if (K == 128)
        gemm_scale_wmma<128><<<tiles / 4, 128, 0, stream>>>(xin, W, dis, G, A, N);
    else
        gemm_scale_wmma<64><<<tiles / 4, 128, 0, stream>>>(xin, W, dis, G, A, N);

    const int nshift = ilog2i(N);
    const long long etot = E << nshift;
    edge_scatter<<<(int)((etot + 255) / 256), 256, 0, stream>>>(src, dst, E, nshift, G, A);

    const long long ftot = (long long)N_NODES << nshift;
    finalize_relu<<<(int)((ftot + 255) / 256), 256, 0, stream>>>(A, dis, bia, out, nshift, ftot);
}

extern "C" void kernel_launch(void* const* d_in, const int* in_sizes, int n_in,
                              void* d_out, int out_size, void* d_ws, size_t ws_size,
                              hipStream_t stream) {
    // Input order: x, edge_index, Wm1,bm1,Wm2,bm2,Wm3,bm3, Wl1,bl1,Wl2,bl2,Wl3,bl3
    const float* x  = (const float*)d_in[0];
    const int*   ei = (const int*)d_in[1];       // integer inputs arrive as int32
    const long long E = in_sizes[1] / 2;
    const int* src = ei;
    const int* dst = ei + E;

    // workspace carve-up (floats): dis | G | A | H   (~77 MB total)
    float* dis = (float*)d_ws;
    float* G   = dis + 50176;                       // padded, 128-aligned
    float* A   = G + (size_t)N_NODES * 128;
    float* H   = A + (size_t)N_NODES * 128;
    float* out = (float*)d_out;

    // normalization: dis = rsqrt(deg + 1)
    zero_f32<<<(N_NODES + 255) / 256, 256, 0, stream>>>(dis, N_NODES);
    deg_count<<<(int)((E + 255) / 256), 256, 0, stream>>>(dst, (int)E, dis);
    make_dis<<<(N_NODES + 255) / 256, 256, 0, stream>>>(dis, N_NODES);

    const int din = in_sizes[0] / N_NODES;          // 128
    for (int s = 0; s < 2; ++s) {                   // s=0: mu stack, s=1: log stack
        const int base = 2 + s * 6;
        const float* W1 = (const float*)d_in[base + 0];
        const float* b1 = (const float*)d_in[base + 1];
        const float* W2 = (const float*)d_in[base + 2];
        const float* b2 = (const float*)d_in[base + 3];
        const float* W3 = (const float*)d_in[base + 4];
        const float* b3 = (const float*)d_in[base + 5];
        const int d1 = in_sizes[base + 1];          // 64
        const int d2 = in_sizes[base + 3];          // 128
        const int d3 = in_sizes[base + 5];          // 64 (latent)

        float* out_s = out + (size_t)s * (size_t)N_NODES * (size_t)d3;

        run_layer(x, W1, b1, din, d1, src, dst, E, dis, G, A, H,     stream);
        run_layer(H, W2, b2, d1,  d2, src, dst, E, dis, G, A, H,     stream); // in-place safe: gemm consumed H before finalize writes it
        run_layer(H, W3, b3, d2,  d3, src, dst, E, dis, G, A, out_s, stream);
    }
}